// GNNActor_21534966022295
// MI455X (gfx1250) — compile-verified
//
#include <hip/hip_runtime.h>

typedef float v2f __attribute__((ext_vector_type(2)));
typedef float v8f __attribute__((ext_vector_type(8)));

#define N_NODES 50000
#define N_EDGES 800000
#define NB (N_NODES / 16)      // 3125 row-tiles, exact
#define TPITCH 132             // padded LDS row pitch (floats) -> conflict-free frag reads

__device__ __forceinline__ v8f wmma_f32(v2f a, v2f b, v8f c) {
  // V_WMMA_F32_16X16X4_F32 : D = A(16x4) * B(4x16) + C(16x16), fp32 throughout
  return __builtin_amdgcn_wmma_f32_16x16x4_f32(false, a, false, b, (short)0, c,
                                               false, false);
}

// ---------------- init: deg = 1 (self loop), agg = 0 ----------------
__global__ void k_init(float* __restrict__ deg, float* __restrict__ agg) {
  int i = blockIdx.x * 256 + threadIdx.x;
  if (i < N_NODES * 128) agg[i] = 0.0f;
  if (i < N_NODES) deg[i] = 1.0f;
}

// ---------------- degree accumulation over dst ----------------
__global__ void k_deg(const int* __restrict__ ei, float* __restrict__ deg) {
  int e = blockIdx.x * 256 + threadIdx.x;
  if (e < N_EDGES) atomicAdd(&deg[ei[N_EDGES + e]], 1.0f);
}

// ---------------- dinv = rsqrt(deg) ----------------
__global__ void k_dinv(const float* __restrict__ deg, float* __restrict__ dinv) {
  int i = blockIdx.x * 256 + threadIdx.x;
  if (i < N_NODES) {
    float d = deg[i];
    dinv[i] = d > 0.0f ? rsqrtf(d) : 0.0f;
  }
}

// ---------------- xw = x @ Wg  (WMMA f32 16x16x4) ----------------
__global__ void k_gemm1(const float* __restrict__ x, const float* __restrict__ Wg,
                        float* __restrict__ xw) {
  __shared__ float lds[4 * 16 * TPITCH];
  const int lane = threadIdx.x & 31;
  const int wave = threadIdx.x >> 5;
  const int half = lane >> 4;   // which 16-lane half
  const int lm   = lane & 15;   // M (A-frag) / N-in-tile (B,D-frag)
  float* tile = lds + wave * (16 * TPITCH);
  const int gw = blockIdx.x * 4 + wave;
  const int nwaves = gridDim.x * 4;

  for (int blk = gw; blk < NB; blk += nwaves) {
    const int row0 = blk * 16;
    // stage 16x128 x-tile into LDS (coalesced float4 per row)
    for (int r = 0; r < 16; ++r) {
      float4 v = *(const float4*)(x + (size_t)(row0 + r) * 128 + lane * 4);
      *(float4*)(tile + r * TPITCH + lane * 4) = v;
    }
    asm volatile("s_wait_dscnt 0" ::: "memory");  // wave-local LDS visibility

    v8f acc[8];
    for (int t = 0; t < 8; ++t) acc[t] = (v8f){};
    for (int kk = 0; kk < 32; ++kk) {
      const int k0 = 4 * kk + 2 * half;
      v2f a = *(const v2f*)(tile + lm * TPITCH + k0);   // A: (M=lm, K=k0..k0+1)
      for (int t = 0; t < 8; ++t) {
        const float* bp = Wg + (size_t)k0 * 128 + 16 * t + lm;
        v2f b; b.x = bp[0]; b.y = bp[128];              // B: (K, N=16t+lm)
        acc[t] = wmma_f32(a, b, acc[t]);
      }
    }
    for (int t = 0; t < 8; ++t) {
      float* op = xw + (size_t)(row0 + 8 * half) * 128 + 16 * t + lm;
      for (int r = 0; r < 8; ++r) op[(size_t)r * 128] = acc[t][r];
    }
    asm volatile("s_wait_dscnt 0" ::: "memory");
  }
}

// ---------------- edge scatter: agg[dst] += xw[src] * dinv[src]*dinv[dst] ----
// one wave per edge, each lane handles 4 consecutive floats (coalesced)
__global__ void k_scatter(const int* __restrict__ ei, const float* __restrict__ dinv,
                          const float* __restrict__ xw, float* __restrict__ agg) {
  int t = blockIdx.x * 256 + threadIdx.x;
  int e = t >> 5;
  int c = (t & 31) << 2;
  if (e >= N_EDGES) return;
  int s = ei[e];
  int d = ei[N_EDGES + e];
  float nrm = dinv[s] * dinv[d];
  float4 v = *(const float4*)(xw + (size_t)s * 128 + c);
  float* a = agg + (size_t)d * 128 + c;
  atomicAdd(a + 0, v.x * nrm);
  atomicAdd(a + 1, v.y * nrm);
  atomicAdd(a + 2, v.z * nrm);
  atomicAdd(a + 3, v.w * nrm);
}

// ---------------- fused: h = relu(agg + selfloop + bg) + x ; 3-layer MLP ------
__global__ void k_mlp(const float* __restrict__ x, const float* __restrict__ xw,
                      const float* __restrict__ agg, const float* __restrict__ dinv,
                      const float* __restrict__ bg,
                      const float* __restrict__ W1, const float* __restrict__ b1,
                      const float* __restrict__ W2, const float* __restrict__ b2,
                      const float* __restrict__ W3, const float* __restrict__ b3,
                      float* __restrict__ out) {
  __shared__ float lds[4 * 16 * TPITCH];
  const int lane = threadIdx.x & 31;
  const int wave = threadIdx.x >> 5;
  const int half = lane >> 4;
  const int lm   = lane & 15;
  float* tile = lds + wave * (16 * TPITCH);
  const float4 bgv = *(const float4*)(bg + lane * 4);
  const int gw = blockIdx.x * 4 + wave;
  const int nwaves = gridDim.x * 4;

  for (int blk = gw; blk < NB; blk += nwaves) {
    const int row0 = blk * 16;
    // ---- build h tile in LDS: relu(agg + dinv^2*xw + bg) + x ----
    for (int r = 0; r < 16; ++r) {
      const int row = row0 + r;
      float di = dinv[row];
      float sl = di * di;  // self-loop norm
      float4 ag  = *(const float4*)(agg + (size_t)row * 128 + lane * 4);
      float4 xwv = *(const float4*)(xw  + (size_t)row * 128 + lane * 4);
      float4 xv  = *(const float4*)(x   + (size_t)row * 128 + lane * 4);
      float4 h;
      h.x = fmaxf(ag.x + xwv.x * sl + bgv.x, 0.0f) + xv.x;
      h.y = fmaxf(ag.y + xwv.y * sl + bgv.y, 0.0f) + xv.y;
      h.z = fmaxf(ag.z + xwv.z * sl + bgv.z, 0.0f) + xv.z;
      h.w = fmaxf(ag.w + xwv.w * sl + bgv.w, 0.0f) + xv.w;
      *(float4*)(tile + r * TPITCH + lane * 4) = h;
    }
    asm volatile("s_wait_dscnt 0" ::: "memory");

    // ---- layer 1: h(16x128) @ W1(128x128) ----
    v8f acc1[8];
    for (int t = 0; t < 8; ++t) acc1[t] = (v8f){};
    for (int kk = 0; kk < 32; ++kk) {
      const int k0 = 4 * kk + 2 * half;
      v2f a = *(const v2f*)(tile + lm * TPITCH + k0);
      for (int t = 0; t < 8; ++t) {
        const float* bp = W1 + (size_t)k0 * 128 + 16 * t + lm;
        v2f b; b.x = bp[0]; b.y = bp[128];
        acc1[t] = wmma_f32(a, b, acc1[t]);
      }
    }
    asm volatile("s_wait_dscnt 0" ::: "memory");
    for (int t = 0; t < 8; ++t) {                 // bias + relu, D-layout -> LDS
      float bb = b1[16 * t + lm];
      for (int r = 0; r < 8; ++r)
        tile[(r + 8 * half) * TPITCH + 16 * t + lm] = fmaxf(acc1[t][r] + bb, 0.0f);
    }
    asm volatile("s_wait_dscnt 0" ::: "memory");

    // ---- layer 2: h1(16x128) @ W2(128x64) ----
    v8f acc2[4];
    for (int t = 0; t < 4; ++t) acc2[t] = (v8f){};
    for (int kk = 0; kk < 32; ++kk) {
      const int k0 = 4 * kk + 2 * half;
      v2f a = *(const v2f*)(tile + lm * TPITCH + k0);
      for (int t = 0; t < 4; ++t) {
        const float* bp = W2 + (size_t)k0 * 64 + 16 * t + lm;
        v2f b; b.x = bp[0]; b.y = bp[64];
        acc2[t] = wmma_f32(a, b, acc2[t]);
      }
    }
    asm volatile("s_wait_dscnt 0" ::: "memory");
    for (int t = 0; t < 4; ++t) {
      float bb = b2[16 * t + lm];
      for (int r = 0; r < 8; ++r)
        tile[(r + 8 * half) * TPITCH + 16 * t + lm] = fmaxf(acc2[t][r] + bb, 0.0f);
    }
    asm volatile("s_wait_dscnt 0" ::: "memory");

    // ---- layer 3: h2(16x64) @ W3(64x2), B padded to 16 cols with zeros ----
    v8f acc3 = (v8f){};
    for (int kk = 0; kk < 16; ++kk) {
      const int k0 = 4 * kk + 2 * half;
      v2f a = *(const v2f*)(tile + lm * TPITCH + k0);
      v2f b;
      if (lm < 2) { b.x = W3[k0 * 2 + lm]; b.y = W3[(k0 + 1) * 2 + lm]; }
      else        { b.x = 0.0f;            b.y = 0.0f; }
      acc3 = wmma_f32(a, b, acc3);
    }
    if (lm < 2) {
      float bb = b3[lm];
      for (int r = 0; r < 8; ++r)
        out[(size_t)lm * N_NODES + row0 + r + 8 * half] = acc3[r] + bb;
    }
  }
}

extern "C" void kernel_launch(void* const* d_in, const int* in_sizes, int n_in,
                              void* d_out, int out_size, void* d_ws, size_t ws_size,
                              hipStream_t stream) {
  (void)in_sizes; (void)n_in; (void)out_size; (void)ws_size;
  const float* x  = (const float*)d_in[0];
  const int*   ei = (const int*)d_in[1];
  const float* Wg = (const float*)d_in[2];
  const float* bg = (const float*)d_in[3];
  const float* W1 = (const float*)d_in[4];
  const float* b1 = (const float*)d_in[5];
  const float* W2 = (const float*)d_in[6];
  const float* b2 = (const float*)d_in[7];
  const float* W3 = (const float*)d_in[8];
  const float* b3 = (const float*)d_in[9];
  float* out = (float*)d_out;

  float* ws   = (float*)d_ws;
  float* deg  = ws;
  float* dinv = ws + N_NODES;
  float* xw   = ws + 2 * N_NODES;
  float* agg  = xw + (size_t)N_NODES * 128;

  k_init<<<(N_NODES * 128 + 255) / 256, 256, 0, stream>>>(deg, agg);
  k_deg<<<(N_EDGES + 255) / 256, 256, 0, stream>>>(ei, deg);
  k_dinv<<<(N_NODES + 255) / 256, 256, 0, stream>>>(deg, dinv);
  k_gemm1<<<782, 128, 0, stream>>>(x, Wg, xw);
  k_scatter<<<(N_EDGES * 32) / 256, 256, 0, stream>>>(ei, dinv, xw, agg);
  k_mlp<<<782, 128, 0, stream>>>(x, xw, agg, dinv, bg, W1, b1, W2, b2, W3, b3, out);
}